// EMRouting_27487790695238
// MI455X (gfx1250) — compile-verified
//
#include <hip/hip_runtime.h>

typedef __attribute__((ext_vector_type(2))) float v2f;
typedef __attribute__((ext_vector_type(4))) float v4f;
typedef __attribute__((ext_vector_type(8))) float v8f;

#define SITES    576          // B*H*W = 4*12*12
#define KTOT     288          // KH*KW*C_in = 3*3*32
#define COMP     512          // C_out * 16 poses
#define MU_ELEMS (SITES * COMP)   // 294912 floats of mu output
#define INV_TEMP 0.18549375f  // 1 - 0.95^4
#define MIN_VAR  5e-4f

__global__ __launch_bounds__(512) void em_routing_kernel(
    const float* __restrict__ votes,
    const float* __restrict__ acts,
    const float* __restrict__ beta_a,
    const float* __restrict__ beta_u,
    float* __restrict__ out)
{
    __shared__ float wsh[KTOT];        // w[k] = a^3
    __shared__ float red1[4 * COMP];   // split-K partials of S1
    __shared__ float red2[4 * COMP];   // split-K partials of S2
    __shared__ float hsh[COMP];        // log(sigma) per component
    __shared__ float Lsh[32];          // sum over poses per cout
    __shared__ float srSh;

    const int s       = blockIdx.x;
    const int tid     = threadIdx.x;
    const int g       = tid >> 7;      // k-group 0..3
    const int lane128 = tid & 127;     // owns components 4*lane128 .. +3

    const v4f* __restrict__ V4 =
        (const v4f*)(votes + (size_t)s * ((size_t)KTOT * COMP));

    // Kick the HBM stream early (overlaps with serial Sr setup below).
    __builtin_prefetch(&V4[g * 128 + lane128], 0, 0);

    // Phase 0: w = a^3, Sr = sum(w)
    if (tid < KTOT) {
        float a = acts[(size_t)s * KTOT + tid];
        wsh[tid] = a * a * a;
    }
    __syncthreads();
    if (tid == 0) {
        float t = 0.0f;
        for (int i = 0; i < KTOT; ++i) t += wsh[i];
        srSh = t;
    }
    __syncthreads();

    // Phase 1: stream V once (non-temporal b128), one-pass moments.
    float s1[4] = {0.f, 0.f, 0.f, 0.f};
    float s2[4] = {0.f, 0.f, 0.f, 0.f};
    #pragma unroll 4
    for (int k = g; k < KTOT; k += 4) {
        float wk = wsh[k];
        v4f v = __builtin_nontemporal_load(&V4[k * 128 + lane128]);
        float t0 = wk * v.x, t1 = wk * v.y, t2 = wk * v.z, t3 = wk * v.w;
        s1[0] += t0; s1[1] += t1; s1[2] += t2; s1[3] += t3;
        s2[0] = fmaf(t0, v.x, s2[0]);
        s2[1] = fmaf(t1, v.y, s2[1]);
        s2[2] = fmaf(t2, v.z, s2[2]);
        s2[3] = fmaf(t3, v.w, s2[3]);
    }
    {
        int b = g * COMP + 4 * lane128;
        red1[b + 0] = s1[0]; red1[b + 1] = s1[1];
        red1[b + 2] = s1[2]; red1[b + 3] = s1[3];
        red2[b + 0] = s2[0]; red2[b + 1] = s2[1];
        red2[b + 2] = s2[2]; red2[b + 3] = s2[3];
    }
    __syncthreads();

    // Phase 2: combine split-K, finalize mu / log(sigma). Thread tid owns component tid.
    const float Sr = srSh;
    const int   c  = tid;
    float S1 = red1[c] + red1[COMP + c] + red1[2 * COMP + c] + red1[3 * COMP + c];
    float S2 = red2[c] + red2[COMP + c] + red2[2 * COMP + c] + red2[3 * COMP + c];
    float mu   = S1 / Sr + 1e-7f;
    float sig2 = (S2 - 2.0f * mu * S1 + mu * mu * Sr) / Sr + MIN_VAR;
    out[(size_t)s * COMP + c] = mu;
    hsh[c] = 0.5f * __logf(sig2);   // log(sigma)
    __syncthreads();

    // Phase 3: L[cout] = sum over 16 poses of log(sigma), via WMMA on wave 0.
    // D = A x Ones + C with A[m,k] = h[cout=base+m, pose=4j+k], 4 accumulating
    // 16x16x4 WMMAs per 16-cout half. EXEC is all-ones inside wave 0.
    if (tid < 32) {
        const int m    = tid & 15;
        const int koff = (tid < 16) ? 0 : 2;   // 16x4 f32 A layout (ISA 7.12.2)
        #pragma unroll
        for (int half = 0; half < 2; ++half) {
            const int base = half << 4;
            v8f acc = {};
            #pragma unroll
            for (int j = 0; j < 4; ++j) {
                v2f a;
                a.x = hsh[(base + m) * 16 + 4 * j + koff];
                a.y = hsh[(base + m) * 16 + 4 * j + koff + 1];
                v2f bv; bv.x = 1.0f; bv.y = 1.0f;
                acc = __builtin_amdgcn_wmma_f32_16x16x4_f32(
                        false, a, false, bv, (short)0, acc, false, false);
            }
            // D row r in VGPR r: lanes 0-15 -> M=r, lanes 16-31 -> M=8+r.
            if (tid == 0) {
                #pragma unroll
                for (int r = 0; r < 8; ++r) Lsh[base + r] = acc[r];
            }
            if (tid == 16) {
                #pragma unroll
                for (int r = 0; r < 8; ++r) Lsh[base + 8 + r] = acc[r];
            }
        }
    }
    __syncthreads();

    // Phase 4: a_j[c4, cout] = sigmoid(inv_temp * (beta_a[c4] - Sr*(16*beta_u[c4] + L[cout])))
    float* aout = out + MU_ELEMS + (size_t)s * 1024;
    {
        int q0 = tid, q1 = tid + 512;
        int c40 = q0 >> 5, co0 = q0 & 31;
        int c41 = q1 >> 5, co1 = q1 & 31;
        float x0 = INV_TEMP * (beta_a[c40] - Sr * (16.0f * beta_u[c40] + Lsh[co0]));
        float x1 = INV_TEMP * (beta_a[c41] - Sr * (16.0f * beta_u[c41] + Lsh[co1]));
        aout[q0] = 1.0f / (1.0f + __expf(-x0));
        aout[q1] = 1.0f / (1.0f + __expf(-x1));
    }
}

extern "C" void kernel_launch(void* const* d_in, const int* in_sizes, int n_in,
                              void* d_out, int out_size, void* d_ws, size_t ws_size,
                              hipStream_t stream) {
    const float* votes = (const float*)d_in[0];
    const float* acts  = (const float*)d_in[1];
    const float* ba    = (const float*)d_in[2];
    const float* bu    = (const float*)d_in[3];
    float* out = (float*)d_out;

    em_routing_kernel<<<SITES, 512, 0, stream>>>(votes, acts, ba, bu, out);

    (void)in_sizes; (void)n_in; (void)out_size; (void)d_ws; (void)ws_size;
}